// ObservedLearnedSinkhornBase2_50036368998776
// MI455X (gfx1250) — compile-verified
//
#include <hip/hip_runtime.h>
#include <math.h>

#define LOG2E 1.4426950408889634f
#define LN2   0.6931471805599453f

constexpr int B_ = 4, M_ = 2048, N_ = 2048, D_ = 256, P_ = 8, H_ = 64, F_ = 16;
constexpr int MP1 = M_ + 1, NP1 = N_ + 1;
constexpr float NORM_  = -12.0f;   // -log2(M+N) = -log2(4096)
constexpr float LOG2N_ = 11.0f;    // log2(2048)
constexpr float LOG2M_ = 11.0f;

typedef __attribute__((ext_vector_type(2))) float v2f;
typedef __attribute__((ext_vector_type(8))) float v8f;

__device__ __forceinline__ float exp2_fast(float x) { return __builtin_amdgcn_exp2f(x); }
__device__ __forceinline__ float log2_fast(float x) { return __builtin_amdgcn_logf(x); }
__device__ __forceinline__ float gelu_exact(float x) {
  return 0.5f * x * (1.0f + erff(x * 0.7071067811865476f));
}
__device__ __forceinline__ float wave_max32(float x) {
#pragma unroll
  for (int m = 16; m >= 1; m >>= 1) x = fmaxf(x, __shfl_xor(x, m, 32));
  return x;
}
__device__ __forceinline__ float wave_sum32(float x) {
#pragma unroll
  for (int m = 16; m >= 1; m >>= 1) x += __shfl_xor(x, m, 32);
  return x;
}
// reduce within 16-lane halves (rows stay in their half in WMMA C layout)
__device__ __forceinline__ float half_sum16(float x) {
#pragma unroll
  for (int m = 8; m >= 1; m >>= 1) x += __shfl_xor(x, m, 32);
  return x;
}
// branchless sorted top-5 insert (descending t0..t4)
__device__ __forceinline__ void top5_insert(float x, float& t0, float& t1, float& t2,
                                            float& t3, float& t4) {
  float m;
  m = fmaxf(t0, x); x = fminf(t0, x); t0 = m;
  m = fmaxf(t1, x); x = fminf(t1, x); t1 = m;
  m = fmaxf(t2, x); x = fminf(t2, x); t2 = m;
  m = fmaxf(t3, x); x = fminf(t3, x); t3 = m;
  t4 = fmaxf(t4, x);
}

// ---------------------------------------------------------------- init u=v=0, dustbin desc rows = 0
__global__ void k_init(float* u, float* v, float* dA, float* dB) {
  int tid = blockIdx.x * blockDim.x + threadIdx.x;
  if (tid < B_ * MP1) { u[tid] = 0.f; v[tid] = 0.f; }
  if (tid < B_ * P_) {
    int b = tid / P_, p = tid % P_;
    dA[((size_t)b * MP1 + M_) * P_ + p] = 0.f;
    dB[((size_t)b * NP1 + N_) * P_ + p] = 0.f;
  }
}

// ---------------------------------------------------------------- descriptor projection via WMMA f32 16x16x4
// dOut[b,m,p] = sum_d mdesc[b,d,m] * pw[p,d] + pb[p] ; one wave per 16-row tile, K=256 -> 64 WMMAs.
// No guarded loads: B columns 8..15 come from a clamped row times a 0 mask.
__global__ void k_proj(const float* __restrict__ mdesc, const float* __restrict__ pw,
                       const float* __restrict__ pb, float* __restrict__ dOut) {
  constexpr int MLEN = M_;  // M_ == N_ == 2048 (compile-time stride -> immediate offsets)
  int lane = threadIdx.x;
  int rl = lane & 15, hi = lane >> 4;
  int b = blockIdx.y;
  int m0 = blockIdx.x * 16;
  // lane-fixed bases: A element (row=rl, k=d0+2*hi(+1)); B element (col=rl, k=d0+2*hi(+1))
  const float* pa = mdesc + (size_t)b * D_ * MLEN + (size_t)(2 * hi) * MLEN + m0 + rl;
  int nc = (rl < P_) ? rl : (P_ - 1);
  float nmask = (rl < P_) ? 1.f : 0.f;
  const float* pwb = pw + nc * D_ + 2 * hi;
  v8f c = {};
#pragma unroll
  for (int d0 = 0; d0 < D_; d0 += 4) {
    v2f a, bv;
    a.x = pa[(size_t)d0 * MLEN];
    a.y = pa[(size_t)(d0 + 1) * MLEN];
    bv.x = pwb[d0] * nmask;
    bv.y = pwb[d0 + 1] * nmask;
    c = __builtin_amdgcn_wmma_f32_16x16x4_f32(false, a, false, bv, (short)0, c, false, false);
  }
  if (rl < P_) {
    float bias = pb[rl];
#pragma unroll
    for (int r = 0; r < 8; ++r) {
      int mrow = m0 + r + 8 * hi; // C layout: VGPR r holds rows r (lanes 0-15) / r+8 (lanes 16-31)
      dOut[((size_t)b * (MLEN + 1) + mrow) * P_ + rl] = c[r] + bias;
    }
  }
}

// ---------------------------------------------------------------- row half-iteration: one wave per row
// u_pre = log2_mu - l2se(S+v) and the 16 features; exploits lse(lt)=log2_mu and the u-cancel in gaps.
__global__ void k_row_pass(const float* __restrict__ scores, const float* __restrict__ alphaP,
                           const float* __restrict__ v, const float* __restrict__ dA,
                           float* __restrict__ u, float* __restrict__ fr) {
  __shared__ __align__(16) float sv[NP1 + 3];
  int b = blockIdx.y;
  for (int idx = threadIdx.x; idx < NP1; idx += blockDim.x) sv[idx] = v[b * NP1 + idx];
  __syncthreads();
  int warp = threadIdx.x >> 5, lane = threadIdx.x & 31;
  int i = blockIdx.x * 8 + warp;
  if (i >= MP1) return;
  float aL = alphaP[0] * LOG2E;
  bool realRow = i < M_;                        // wave-uniform
  const float4* srow4 = (const float4*)(scores + ((size_t)b * M_ + (realRow ? i : 0)) * N_);
  const float4* sv4 = (const float4*)sv;
  const float NEG = -__builtin_inff();
  // pass 1: per-lane top-5 over the 2048 real columns (float4 = global_load_b128)
  float t0 = NEG, t1 = NEG, t2 = NEG, t3 = NEG, t4 = NEG;
  for (int it = 0; it < N_ / 128; ++it) {
    int q = it * 32 + lane;
    __builtin_prefetch(srow4 + q + 64, 0, 0);
    float4 sc = srow4[q];
    float4 vv = sv4[q];
    top5_insert((realRow ? sc.x * LOG2E : aL) + vv.x, t0, t1, t2, t3, t4);
    top5_insert((realRow ? sc.y * LOG2E : aL) + vv.y, t0, t1, t2, t3, t4);
    top5_insert((realRow ? sc.z * LOG2E : aL) + vv.z, t0, t1, t2, t3, t4);
    top5_insert((realRow ? sc.w * LOG2E : aL) + vv.w, t0, t1, t2, t3, t4);
  }
  // dustbin column j=N: insert exactly once (lane 0 only)
  if (lane == 0) top5_insert(aL + sv[N_], t0, t1, t2, t3, t4);
  // cross-lane top-5: 5 rounds of wave-max + retire-one (first matching lane pops)
  float top[5];
  int p = 0;
#pragma unroll
  for (int k = 0; k < 5; ++k) {
    float cand = (p == 0) ? t0 : (p == 1) ? t1 : (p == 2) ? t2 : (p == 3) ? t3 : (p == 4) ? t4 : NEG;
    float wm = wave_max32(cand);
    unsigned long long bal = __ballot(cand == wm);
    int first = __ffsll(bal) - 1;
    if (lane == first) p++;
    top[k] = wm;
  }
  float gmax = top[0];
  // pass 2: one exp2 per element (L2-hot re-read)
  float acc = 0.f;
  for (int it = 0; it < N_ / 128; ++it) {
    int q = it * 32 + lane;
    float4 sc = srow4[q];
    float4 vv = sv4[q];
    acc += exp2_fast((realRow ? sc.x * LOG2E : aL) + vv.x - gmax);
    acc += exp2_fast((realRow ? sc.y * LOG2E : aL) + vv.y - gmax);
    acc += exp2_fast((realRow ? sc.z * LOG2E : aL) + vv.z - gmax);
    acc += exp2_fast((realRow ? sc.w * LOG2E : aL) + vv.w - gmax);
  }
  if (lane == 0) acc += exp2_fast(aL + sv[N_] - gmax);
  acc = wave_sum32(acc);
  float lse = gmax + log2_fast(acc);
  float log2mu = realRow ? NORM_ : (NORM_ + LOG2N_);
  float upre = log2mu - lse;
  if (lane == 0) u[b * MP1 + i] = upre;
  if (lane < 16) {
    float feat;
    if (lane == 0)      feat = log2mu * LN2;                 // lse(lt) == log2_mu
    else if (lane == 1) feat = upre * LN2;                   // offset
    else if (lane < 7)  feat = (gmax - top[lane - 2]) * LN2; // gaps (u cancels)
    else if (lane == 7) feat = (lse - gmax) * LN2;           // spread
    else feat = dA[((size_t)b * MP1 + i) * P_ + (lane - 8)];
    fr[((size_t)b * MP1 + i) * F_ + lane] = feat;
  }
}

// ---------------------------------------------------------------- column half-iteration: one thread per column
__global__ void k_col_pass(const float* __restrict__ scores, const float* __restrict__ alphaP,
                           const float* __restrict__ u, const float* __restrict__ dB,
                           float* __restrict__ v, float* __restrict__ fc) {
  __shared__ float su[MP1];
  int b = blockIdx.y;
  for (int idx = threadIdx.x; idx < MP1; idx += blockDim.x) su[idx] = u[b * MP1 + idx];
  __syncthreads();
  int j = blockIdx.x * blockDim.x + threadIdx.x;
  if (j >= NP1) return;
  float aL = alphaP[0] * LOG2E;
  bool realCol = j < N_;
  int jc = realCol ? j : (N_ - 1);              // clamped -> unconditional coalesced loads
  const float* scol = scores + (size_t)b * M_ * N_ + jc;
  const float NEG = -__builtin_inff();
  float t0 = NEG, t1 = NEG, t2 = NEG, t3 = NEG, t4 = NEG;
#pragma unroll 4
  for (int i = 0; i < M_; ++i) {
    float sc = scol[(size_t)i * N_] * LOG2E;
    float x = (realCol ? sc : aL) + su[i];
    top5_insert(x, t0, t1, t2, t3, t4);
  }
  top5_insert(aL + su[M_], t0, t1, t2, t3, t4); // dustbin row i=M
  float gmax = t0, acc = 0.f;
#pragma unroll 4
  for (int i = 0; i < M_; ++i) {
    float sc = scol[(size_t)i * N_] * LOG2E;
    acc += exp2_fast((realCol ? sc : aL) + su[i] - gmax);
  }
  acc += exp2_fast(aL + su[M_] - gmax);
  float lse = gmax + log2_fast(acc);
  float log2nu = realCol ? NORM_ : (NORM_ + LOG2M_);
  float vpre = log2nu - lse;
  v[b * NP1 + j] = vpre;
  float* fp = fc + ((size_t)b * NP1 + j) * F_;
  fp[0] = log2nu * LN2;
  fp[1] = vpre * LN2;
  fp[2] = 0.f;
  fp[3] = (gmax - t1) * LN2;
  fp[4] = (gmax - t2) * LN2;
  fp[5] = (gmax - t3) * LN2;
  fp[6] = (gmax - t4) * LN2;
  fp[7] = (lse - gmax) * LN2;
#pragma unroll
  for (int pp = 0; pp < P_; ++pp) fp[8 + pp] = dB[((size_t)b * NP1 + j) * P_ + pp];
}

// ---------------------------------------------------------------- WMMA MLP 16->64->64->1 over 16-row tiles
__global__ void k_mlp(const float* __restrict__ fr,
                      const float* __restrict__ w1, const float* __restrict__ b1,
                      const float* __restrict__ g1, const float* __restrict__ be1,
                      const float* __restrict__ w2, const float* __restrict__ b2,
                      const float* __restrict__ g2, const float* __restrict__ be2,
                      const float* __restrict__ w3, const float* __restrict__ b3,
                      float* __restrict__ uv, int rows) {
  __shared__ float lds_h[16 * 64];
  int lane = threadIdx.x;
  int rl = lane & 15, hi = lane >> 4;
  int b = blockIdx.y;
  int row0 = blockIdx.x * 16;
  const float* frb = fr + (size_t)b * rows * F_;
  int rA = row0 + rl;
  float maskA = (rA < rows) ? 1.f : 0.f;
  int rcl = (rA < rows) ? rA : (rows - 1);
  const float* frrow = frb + (size_t)rcl * F_;

  // ---- layer 1: (16x16) @ (16x64), K=16 -> 4 WMMA steps x 4 N-tiles
  v8f acc[4] = {};
#pragma unroll
  for (int s = 0; s < 4; ++s) {
    int kb = 4 * s + 2 * hi;
    v2f a;
    a.x = frrow[kb] * maskA;
    a.y = frrow[kb + 1] * maskA;
#pragma unroll
    for (int nt = 0; nt < 4; ++nt) {
      int n = nt * 16 + rl;
      v2f bv;
      bv.x = w1[n * F_ + kb];
      bv.y = w1[n * F_ + kb + 1];
      acc[nt] = __builtin_amdgcn_wmma_f32_16x16x4_f32(false, a, false, bv, (short)0, acc[nt], false, false);
    }
  }
  // bias + exact gelu + layernorm (per-row stats via intra-half shuffles)
  float h[8][4];
#pragma unroll
  for (int nt = 0; nt < 4; ++nt) {
    float bb = b1[nt * 16 + rl];
#pragma unroll
    for (int r = 0; r < 8; ++r) h[r][nt] = gelu_exact(acc[nt][r] + bb);
  }
#pragma unroll
  for (int r = 0; r < 8; ++r) {
    float s = h[r][0] + h[r][1] + h[r][2] + h[r][3];
    float q = h[r][0] * h[r][0] + h[r][1] * h[r][1] + h[r][2] * h[r][2] + h[r][3] * h[r][3];
    s = half_sum16(s); q = half_sum16(q);
    float mu = s * (1.f / 64.f);
    float var = q * (1.f / 64.f) - mu * mu;
    float inv = rsqrtf(var + 1e-5f);
#pragma unroll
    for (int nt = 0; nt < 4; ++nt) {
      int n = nt * 16 + rl;
      h[r][nt] = (h[r][nt] - mu) * inv * g1[n] + be1[n];
    }
  }
  // spill to LDS row-major to re-lay into WMMA A format
#pragma unroll
  for (int r = 0; r < 8; ++r)
#pragma unroll
    for (int nt = 0; nt < 4; ++nt)
      lds_h[(r + 8 * hi) * 64 + nt * 16 + rl] = h[r][nt];
  __syncthreads();

  // ---- layer 2: (16x64) @ (64x64), K=64 -> 16 WMMA steps x 4 N-tiles
  v8f c2[4] = {};
#pragma unroll
  for (int s = 0; s < 16; ++s) {
    int kb = 4 * s + 2 * hi;
    v2f a;
    a.x = lds_h[rl * 64 + kb];
    a.y = lds_h[rl * 64 + kb + 1];
#pragma unroll
    for (int nt = 0; nt < 4; ++nt) {
      int n = nt * 16 + rl;
      v2f bv;
      bv.x = w2[n * H_ + kb];
      bv.y = w2[n * H_ + kb + 1];
      c2[nt] = __builtin_amdgcn_wmma_f32_16x16x4_f32(false, a, false, bv, (short)0, c2[nt], false, false);
    }
  }
#pragma unroll
  for (int nt = 0; nt < 4; ++nt) {
    float bb = b2[nt * 16 + rl];
#pragma unroll
    for (int r = 0; r < 8; ++r) h[r][nt] = gelu_exact(c2[nt][r] + bb);
  }
#pragma unroll
  for (int r = 0; r < 8; ++r) {
    float s = h[r][0] + h[r][1] + h[r][2] + h[r][3];
    float q = h[r][0] * h[r][0] + h[r][1] * h[r][1] + h[r][2] * h[r][2] + h[r][3] * h[r][3];
    s = half_sum16(s); q = half_sum16(q);
    float mu = s * (1.f / 64.f);
    float var = q * (1.f / 64.f) - mu * mu;
    float inv = rsqrtf(var + 1e-5f);
#pragma unroll
    for (int nt = 0; nt < 4; ++nt) {
      int n = nt * 16 + rl;
      h[r][nt] = (h[r][nt] - mu) * inv * g2[n] + be2[n];
    }
  }
  // ---- layer 3: dot with w3 + b3, then uv += delta * LOG2E
  float res[8];
#pragma unroll
  for (int r = 0; r < 8; ++r) {
    float pa = h[r][0] * w3[rl] + h[r][1] * w3[16 + rl] + h[r][2] * w3[32 + rl] + h[r][3] * w3[48 + rl];
    res[r] = half_sum16(pa) + b3[0];
  }
  if (rl == 0) {
#pragma unroll
    for (int r = 0; r < 8; ++r) {
      int row = row0 + r + 8 * hi;
      if (row < rows) uv[b * rows + row] += res[r] * LOG2E;
    }
  }
}

// ---------------------------------------------------------------- final: out = S + u + v - norm (S built on the fly)
__global__ void k_final(const float* __restrict__ scores, const float* __restrict__ alphaP,
                        const float* __restrict__ u, const float* __restrict__ v,
                        float* __restrict__ out) {
  int b = blockIdx.z;
  int i = blockIdx.y;
  int j = blockIdx.x * blockDim.x + threadIdx.x;
  if (j >= NP1) return;
  float aL = alphaP[0] * LOG2E;
  int ic = (i < M_) ? i : (M_ - 1);
  int jc = (j < N_) ? j : (N_ - 1);
  float sc = scores[((size_t)b * M_ + ic) * N_ + jc] * LOG2E;  // unconditional load
  float s = (i < M_ && j < N_) ? sc : aL;
  out[((size_t)b * MP1 + i) * NP1 + j] = s + u[b * MP1 + i] + v[b * NP1 + j] - NORM_;
}

// ---------------------------------------------------------------- launch
extern "C" void kernel_launch(void* const* d_in, const int* in_sizes, int n_in,
                              void* d_out, int out_size, void* d_ws, size_t ws_size,
                              hipStream_t stream) {
  const float* scores = (const float*)d_in[0];
  const float* alpha  = (const float*)d_in[1];
  const float* mdesc0 = (const float*)d_in[2];
  const float* mdesc1 = (const float*)d_in[3];
  const float* pA_w = (const float*)d_in[4];
  const float* pA_b = (const float*)d_in[5];
  const float* pB_w = (const float*)d_in[6];
  const float* pB_b = (const float*)d_in[7];
  const float* rW[10]; for (int k = 0; k < 10; ++k) rW[k] = (const float*)d_in[8 + k];
  const float* cW[10]; for (int k = 0; k < 10; ++k) cW[k] = (const float*)d_in[18 + k];

  float* ws = (float*)d_ws;
  float* u  = ws;                         // B*(M+1) = 8196, padded to 8208
  float* v  = ws + 8208;
  float* dA = ws + 16416;                 // B*(M+1)*P = 65568
  float* dB = dA + 65568;
  float* fr = dB + 65568;                 // B*(M+1)*16 = 131136

  k_init<<<dim3((B_ * MP1 + 255) / 256), 256, 0, stream>>>(u, v, dA, dB);
  k_proj<<<dim3(M_ / 16, B_), 32, 0, stream>>>(mdesc0, pA_w, pA_b, dA);
  k_proj<<<dim3(N_ / 16, B_), 32, 0, stream>>>(mdesc1, pB_w, pB_b, dB);

  for (int it = 0; it < 3; ++it) {
    k_row_pass<<<dim3((MP1 + 7) / 8, B_), 256, 0, stream>>>(scores, alpha, v, dA, u, fr);
    k_mlp<<<dim3((MP1 + 15) / 16, B_), 32, 0, stream>>>(fr, rW[0], rW[1], rW[2], rW[3], rW[4], rW[5], rW[6], rW[7], rW[8], rW[9], u, MP1);
    k_col_pass<<<dim3((NP1 + 255) / 256, B_), 256, 0, stream>>>(scores, alpha, u, dB, v, fr);
    k_mlp<<<dim3((NP1 + 15) / 16, B_), 32, 0, stream>>>(fr, cW[0], cW[1], cW[2], cW[3], cW[4], cW[5], cW[6], cW[7], cW[8], cW[9], v, NP1);
  }

  k_final<<<dim3((NP1 + 255) / 256, MP1, B_), 256, 0, stream>>>(scores, alpha, u, v, (float*)d_out);
}